// InfoNCELoss_26912265077430
// MI455X (gfx1250) — compile-verified
//
#include <hip/hip_runtime.h>

// ---- CDNA5 (gfx1250) InfoNCE, fused streaming log-softmax over Z Z^T ----
// fp32 WMMA path: v_wmma_f32_16x16x4_f32, wave32, double-buffered LDS B-stage.

typedef __attribute__((ext_vector_type(2))) float v2f;
typedef __attribute__((ext_vector_type(8))) float v8f;

#define B_ROWS 8192
#define DIM 256
#define NTOT 16384                 // 2*B
#define WAVES_PER_WG 4
#define THREADS (WAVES_PER_WG * 32)
#define ROWS_PER_WG (WAVES_PER_WG * 16)      // 64
#define NUM_WG (NTOT / ROWS_PER_WG)          // 256 blocks
#define CHUNK_COLS 64
#define NUM_CHUNKS (NTOT / CHUNK_COLS)       // 256
#define TILES_PER_CHUNK (CHUNK_COLS / 16)    // 4
#define COL_STRIDE 260             // 256 + 4 pad: b64 LDS reads hit 64 distinct banks

__device__ __forceinline__ void load_chunk(float* __restrict__ dst,
                                           const float* __restrict__ z1,
                                           const float* __restrict__ z2,
                                           int chunk, int tid) {
  // 128 threads load 64 columns x 256 floats (2 threads per column, 128 floats each)
  const int col = tid >> 1;
  const int q   = tid & 1;
  const int gcol = chunk * CHUNK_COLS + col;
  const float* src = (gcol < B_ROWS ? z1 + (size_t)gcol * DIM
                                    : z2 + (size_t)(gcol - B_ROWS) * DIM) + q * 128;
  float* d = dst + col * COL_STRIDE + q * 128;
#pragma unroll
  for (int i = 0; i < 32; ++i) {
    ((float4*)d)[i] = ((const float4*)src)[i];
  }
}

__global__ __launch_bounds__(THREADS, 1)
void infonce_main(const float* __restrict__ z1, const float* __restrict__ z2,
                  const float* __restrict__ plogit_scale, float* __restrict__ partial) {
  __shared__ __align__(16) float ldsb[2][CHUNK_COLS * COL_STRIDE];  // ~133 KB of 320 KB WGP LDS
  __shared__ float wsum[WAVES_PER_WG];

  const int tid  = threadIdx.x;
  const int wave = tid >> 5;
  const int lane = tid & 31;
  const int half = lane >> 4;   // K-pair selector for fp32 WMMA A/B layout
  const int ln   = lane & 15;   // M (for A) / N (for B, C)

  const float scale = fminf(__expf(plogit_scale[0]), 100.0f);

  const int rt   = blockIdx.x * WAVES_PER_WG + wave;  // this wave's row tile (0..1023)
  const int pt   = rt ^ (B_ROWS / 16);                // tile holding the positive (i <-> i^8192)
  const int grow = rt * 16 + ln;
  const float* zrow = (grow < B_ROWS) ? (z1 + (size_t)grow * DIM)
                                      : (z2 + (size_t)(grow - B_ROWS) * DIM);

  // Preload the 16x256 A-panel into registers: 64 k-steps, v2f each (128 VGPRs).
  // fp32 16x4 A layout: lane holds row M=lane%16, K = ks*4 + 2*(lane/16) + {0,1}.
  v2f afrag[64];
  const float* ap = zrow + 2 * half;
#pragma unroll
  for (int ks = 0; ks < 64; ++ks)
    afrag[ks] = *(const v2f*)(ap + ks * 4);

  // Online-softmax state per lane: 8 C-VGPR slots -> 8 (row, col%16) partials.
  float mrun[8], srun[8], pos[8];
#pragma unroll
  for (int v = 0; v < 8; ++v) { mrun[v] = -3.0e38f; srun[v] = 0.f; pos[v] = 0.f; }

  load_chunk(&ldsb[0][0], z1, z2, 0, tid);
  __syncthreads();

  for (int c = 0; c < NUM_CHUNKS; ++c) {
    const int cur = c & 1;
    if (c + 1 < NUM_CHUNKS) load_chunk(&ldsb[cur ^ 1][0], z1, z2, c + 1, tid);

    for (int t = 0; t < TILES_PER_CHUNK; ++t) {
      const int ct = c * TILES_PER_CHUNK + t;
      // B layout (4x16 fp32): lane holds col N=lane%16, K = ks*4 + 2*(lane/16) + {0,1}
      const float* bp = &ldsb[cur][(t * 16 + ln) * COL_STRIDE + 2 * half];

      v8f acc0 = {}, acc1 = {}, acc2 = {}, acc3 = {};
#pragma unroll
      for (int ks = 0; ks < 64; ks += 4) {
        v2f b0 = *(const v2f*)(bp + (ks + 0) * 4);
        v2f b1 = *(const v2f*)(bp + (ks + 1) * 4);
        v2f b2 = *(const v2f*)(bp + (ks + 2) * 4);
        v2f b3 = *(const v2f*)(bp + (ks + 3) * 4);
        acc0 = __builtin_amdgcn_wmma_f32_16x16x4_f32(false, afrag[ks + 0], false, b0,
                                                     (short)0, acc0, false, false);
        acc1 = __builtin_amdgcn_wmma_f32_16x16x4_f32(false, afrag[ks + 1], false, b1,
                                                     (short)0, acc1, false, false);
        acc2 = __builtin_amdgcn_wmma_f32_16x16x4_f32(false, afrag[ks + 2], false, b2,
                                                     (short)0, acc2, false, false);
        acc3 = __builtin_amdgcn_wmma_f32_16x16x4_f32(false, afrag[ks + 3], false, b3,
                                                     (short)0, acc3, false, false);
      }
      v8f acc = (acc0 + acc1) + (acc2 + acc3);

      const bool dtile = (ct == rt);   // tile containing the masked diagonal
      const bool ptile = (ct == pt);   // tile containing the positive logit
#pragma unroll
      for (int v = 0; v < 8; ++v) {
        // C layout: slot v, lane -> M = v + 8*half, N = lane%16; diag/pos sit at M==N.
        const bool dpos = (ln == v + 8 * half);
        float x = acc[v] * scale;
        if (ptile && dpos) pos[v] += x;
        if (dtile && dpos) x = -1e9f;
        float nm = fmaxf(mrun[v], x);
        srun[v]  = srun[v] * __expf(mrun[v] - nm) + __expf(x - nm);
        mrun[v]  = nm;
      }
    }
    __syncthreads();  // one barrier/iter: protects both buffers (classic double-buffer)
  }

  // Merge (max, sum, pos) across the 16 lanes that share each output row.
  float rsum = 0.f;
#pragma unroll
  for (int v = 0; v < 8; ++v) {
    float m = mrun[v], s = srun[v], p = pos[v];
#pragma unroll
    for (int off = 1; off < 16; off <<= 1) {
      float mo = __shfl_xor(m, off, 32);
      float so = __shfl_xor(s, off, 32);
      float nm = fmaxf(m, mo);
      s = s * __expf(m - nm) + so * __expf(mo - nm);
      m = nm;
      p += __shfl_xor(p, off, 32);
    }
    rsum += (m + __logf(s)) - p;   // nll for row (v + 8*half) of this tile
  }
  rsum += __shfl_xor(rsum, 16, 32);  // add the other half's 8 rows -> wave total
  if (lane == 0) wsum[wave] = rsum;
  __syncthreads();
  if (tid == 0) {
    float s = 0.f;
#pragma unroll
    for (int w = 0; w < WAVES_PER_WG; ++w) s += wsum[w];
    partial[blockIdx.x] = s;
  }
}

__global__ void infonce_finalize(const float* __restrict__ partial, float* __restrict__ out) {
  if (threadIdx.x == 0) {
    float s = 0.f;
    for (int i = 0; i < NUM_WG; ++i) s += partial[i];  // fixed order -> deterministic
    out[0] = s / (float)NTOT;
  }
}

extern "C" void kernel_launch(void* const* d_in, const int* in_sizes, int n_in,
                              void* d_out, int out_size, void* d_ws, size_t ws_size,
                              hipStream_t stream) {
  (void)in_sizes; (void)n_in; (void)out_size; (void)ws_size;
  const float* z1 = (const float*)d_in[0];
  const float* z2 = (const float*)d_in[1];
  const float* ls = (const float*)d_in[2];
  float* partial  = (float*)d_ws;   // 256 floats of scratch

  infonce_main<<<NUM_WG, THREADS, 0, stream>>>(z1, z2, ls, partial);
  infonce_finalize<<<1, 32, 0, stream>>>(partial, (float*)d_out);
}